// PScanTriton_19215683682962
// MI455X (gfx1250) — compile-verified
//
#include <hip/hip_runtime.h>
#include <stdint.h>

// Problem shape (fixed by the reference)
#define Bq 4
#define Lq 2048
#define Cq 1024
#define Pq 16                 // partitions along L
#define Sq (Lq / Pq)          // 128 steps per partition
#define NCHAIN (Bq * Cq)      // 4096 independent (b,c) chains
#define NT 256                // threads per block (8 wave32s)
#define RING 4                // async pipeline depth (3 steps in flight)

// Native 128-bit vector type (works with nontemporal builtins, lowers to b128).
typedef float v4f __attribute__((ext_vector_type(4)));

// 2x2 complex matrix, layout matches reference [i][j][re/im] = 8 floats, 32 B.
struct __align__(16) C22 { float v[8]; };

__device__ __forceinline__ C22 c22_zero() {
  C22 r;
#pragma unroll
  for (int i = 0; i < 8; ++i) r.v[i] = 0.0f;
  return r;
}

__device__ __forceinline__ C22 c22_load(const float* __restrict__ p) {
  C22 r;
  const v4f* q = (const v4f*)p;
  ((v4f*)r.v)[0] = q[0];
  ((v4f*)r.v)[1] = q[1];
  return r;
}

__device__ __forceinline__ void c22_store(float* __restrict__ p, const C22& a) {
  v4f* q = (v4f*)p;
  q[0] = ((const v4f*)a.v)[0];
  q[1] = ((const v4f*)a.v)[1];
}

// Streaming output: non-temporal so the Y stream does not churn L2.
__device__ __forceinline__ void c22_store_nt(float* __restrict__ p, const C22& a) {
  v4f* q = (v4f*)p;
  __builtin_nontemporal_store(((const v4f*)a.v)[0], q);
  __builtin_nontemporal_store(((const v4f*)a.v)[1], q + 1);
}

// complex fused multiply-accumulate: (dr,di) += (ar,ai)*(br,bi)
__device__ __forceinline__ void cfma(float& dr, float& di,
                                     float ar, float ai, float br, float bi) {
  dr = fmaf(ar, br, dr);
  dr = fmaf(-ai, bi, dr);
  di = fmaf(ar, bi, di);
  di = fmaf(ai, br, di);
}

// R = A@B + Cin (2x2 complex)
__device__ __forceinline__ C22 c22_mma(const C22& A, const C22& B, const C22& Cin) {
  C22 R = Cin;
#pragma unroll
  for (int i = 0; i < 2; ++i) {
#pragma unroll
    for (int j = 0; j < 2; ++j) {
      float dr = R.v[i * 4 + j * 2 + 0];
      float di = R.v[i * 4 + j * 2 + 1];
#pragma unroll
      for (int k = 0; k < 2; ++k) {
        cfma(dr, di,
             A.v[i * 4 + k * 2 + 0], A.v[i * 4 + k * 2 + 1],
             B.v[k * 4 + j * 2 + 0], B.v[k * 4 + j * 2 + 1]);
      }
      R.v[i * 4 + j * 2 + 0] = dr;
      R.v[i * 4 + j * 2 + 1] = di;
    }
  }
  return R;
}

__device__ __forceinline__ C22 c22_mm(const C22& A, const C22& B) {
  return c22_mma(A, B, c22_zero());
}

// ---------------- CDNA5 async global->LDS helpers (ASYNCcnt path) -----------
// One 32-byte matrix = two b128 async copies; the +16B leg rides the signed
// 24-bit instruction offset so only one 64-bit address add is needed.
__device__ __forceinline__ void async_ld32(uint32_t lds_lo, uint32_t lds_hi,
                                           const float* gaddr) {
  asm volatile("global_load_async_to_lds_b128 %0, %2, off\n\t"
               "global_load_async_to_lds_b128 %1, %2, off offset:16"
               :: "v"(lds_lo), "v"(lds_hi), "v"(gaddr) : "memory");
}
__device__ __forceinline__ void wait_async_le12() { asm volatile("s_wait_asynccnt 0xc" ::: "memory"); }
__device__ __forceinline__ void wait_async_le8()  { asm volatile("s_wait_asynccnt 0x8" ::: "memory"); }
__device__ __forceinline__ void wait_async_le4()  { asm volatile("s_wait_asynccnt 0x4" ::: "memory"); }
__device__ __forceinline__ void wait_async_le0()  { asm volatile("s_wait_asynccnt 0x0" ::: "memory"); }

// Issue one time-step (A: 32B, X: 32B) = 4 async b128 ops into an LDS slot.
__device__ __forceinline__ void issue_step(uint32_t lds_slot,
                                           const float* Aq, const float* Xq) {
  async_ld32(lds_slot +  0, lds_slot + 16, Aq);
  async_ld32(lds_slot + 32, lds_slot + 48, Xq);
}

// Read one staged step (a, x) out of an LDS slot (per-lane 64 B).
__device__ __forceinline__ void slot_read(const float* sp16, C22& a, C22& x) {
  const v4f* sp = (const v4f*)sp16;
  ((v4f*)a.v)[0] = sp[0];
  ((v4f*)a.v)[1] = sp[1];
  ((v4f*)x.v)[0] = sp[2];
  ((v4f*)x.v)[1] = sp[3];
}

// Common block-index decode: grid = B * P * (C/NT)
__device__ __forceinline__ void decode_blk(int blk, int tid, int& b, int& p, int& c) {
  const int cb = blk & 3;               // C/NT = 4 channel chunks
  p = (blk >> 2) & (Pq - 1);            // partition
  b = blk >> 6;                         // batch
  c = cb * NT + tid;
}

// ---------------------------------------------------------------------------
// Kernel 1: per-(b,c,partition) aggregates (A_agg, X_agg).
// 4-slot async ring keeps 3 steps (12 b128 ops/wave, ~6 KB) in flight; lanes
// map to adjacent channels so each matrix's two async ops cover 1 KB contiguous per wave.
// ---------------------------------------------------------------------------
__global__ __launch_bounds__(NT) void k1_partial(const float* __restrict__ A,
                                                 const float* __restrict__ X,
                                                 float* __restrict__ agg) {
  __shared__ float stage[RING][NT][16];   // 64 KB -> 5 blocks per 320 KB WGP
  const int tid = threadIdx.x;
  int b, p, c;
  decode_blk(blockIdx.x, tid, b, p, c);
  const int l0 = p * Sq;

  const size_t step = (size_t)Cq * 8;     // floats between consecutive l
  const size_t base = (((size_t)b * Lq + l0) * Cq + c) * 8;
  const float* Ap = A + base;
  const float* Xp = X + base;

  const uint32_t lds0 = (uint32_t)(uintptr_t)&stage[0][tid][0];
  const uint32_t slotBytes = NT * 16 * 4; // 16 KB

  // Prologue: steps 0..2 into slots 0..2 (12 async ops outstanding)
#pragma unroll
  for (int s = 0; s < 3; ++s)
    issue_step(lds0 + (uint32_t)s * slotBytes, Ap + s * step, Xp + s * step);

  // Peel i = 0: init aggregate from first element.
  issue_step(lds0 + 3u * slotBytes, Ap + 3 * step, Xp + 3 * step);
  wait_async_le12();
  C22 Ag, Xg;
  slot_read(&stage[0][tid][0], Ag, Xg);

  for (int i = 1; i + 3 < Sq; ++i) {
    issue_step(lds0 + (uint32_t)((i + 3) & 3) * slotBytes,
               Ap + (size_t)(i + 3) * step, Xp + (size_t)(i + 3) * step);
    wait_async_le12();                    // step i's 4 ops complete
    C22 a, x;
    slot_read(&stage[i & 3][tid][0], a, x);
    Ag = c22_mm(a, Ag);                   // A_agg = a @ A_agg
    Xg = c22_mma(a, Xg, x);               // X_agg = a @ X_agg + x
  }
  // Drain: steps S-3, S-2, S-1
  {
    C22 a, x;
    wait_async_le8();
    slot_read(&stage[(Sq - 3) & 3][tid][0], a, x);
    Ag = c22_mm(a, Ag); Xg = c22_mma(a, Xg, x);
    wait_async_le4();
    slot_read(&stage[(Sq - 2) & 3][tid][0], a, x);
    Ag = c22_mm(a, Ag); Xg = c22_mma(a, Xg, x);
    wait_async_le0();
    slot_read(&stage[(Sq - 1) & 3][tid][0], a, x);
    Ag = c22_mm(a, Ag); Xg = c22_mma(a, Xg, x);
  }

  const int chain = b * Cq + c;
  float* dst = agg + ((size_t)p * NCHAIN + chain) * 16;
  c22_store(dst, Ag);
  c22_store(dst + 8, Xg);
}

// ---------------------------------------------------------------------------
// Kernel 2: sequential carry scan over partitions per chain (6 MB, L2-resident).
// yin[p] = exclusive carry entering partition p (zero matrix for p==0).
// ---------------------------------------------------------------------------
__global__ __launch_bounds__(NT) void k2_carry(const float* __restrict__ agg,
                                               float* __restrict__ yin) {
  const int chain = blockIdx.x * NT + threadIdx.x;   // 4096 chains
  C22 carry = c22_zero();
  for (int p = 0; p < Pq; ++p) {
    const float* g = agg + ((size_t)p * NCHAIN + chain) * 16;
    C22 Ag = c22_load(g);
    C22 Xg = c22_load(g + 8);
    c22_store(yin + ((size_t)p * NCHAIN + chain) * 8, carry);
    carry = c22_mma(Ag, carry, Xg);     // C[p] = A_agg @ C[p-1] + X_agg
  }
}

// ---------------------------------------------------------------------------
// Kernel 3: replay each partition from its carry with the same 4-slot async
// ring; outputs streamed with non-temporal b128 stores.
// ---------------------------------------------------------------------------
__global__ __launch_bounds__(NT) void k3_apply(const float* __restrict__ A,
                                               const float* __restrict__ X,
                                               const float* __restrict__ yin,
                                               float* __restrict__ Y) {
  __shared__ float stage[RING][NT][16];   // 64 KB
  const int tid = threadIdx.x;
  int b, p, c;
  decode_blk(blockIdx.x, tid, b, p, c);
  const int chain = b * Cq + c;
  const int l0 = p * Sq;

  const size_t step = (size_t)Cq * 8;
  const size_t base = (((size_t)b * Lq + l0) * Cq + c) * 8;
  const float* Ap = A + base;
  const float* Xp = X + base;
  float* dst = Y + base;

  const uint32_t lds0 = (uint32_t)(uintptr_t)&stage[0][tid][0];
  const uint32_t slotBytes = NT * 16 * 4; // 16 KB

  // Prologue: steps 0..2 into slots 0..2
#pragma unroll
  for (int s = 0; s < 3; ++s)
    issue_step(lds0 + (uint32_t)s * slotBytes, Ap + s * step, Xp + s * step);

  C22 Yprev = c22_load(yin + ((size_t)p * NCHAIN + chain) * 8);

  for (int i = 0; i + 3 < Sq; ++i) {
    issue_step(lds0 + (uint32_t)((i + 3) & 3) * slotBytes,
               Ap + (size_t)(i + 3) * step, Xp + (size_t)(i + 3) * step);
    wait_async_le12();                    // step i's 4 ops complete
    C22 a, x;
    slot_read(&stage[i & 3][tid][0], a, x);
    C22 y = c22_mma(a, Yprev, x);         // Y = a @ Yprev + x
    c22_store_nt(dst + (size_t)i * step, y);
    Yprev = y;
  }
  // Drain: steps S-3, S-2, S-1
  {
    C22 a, x;
    wait_async_le8();
    slot_read(&stage[(Sq - 3) & 3][tid][0], a, x);
    Yprev = c22_mma(a, Yprev, x);
    c22_store_nt(dst + (size_t)(Sq - 3) * step, Yprev);
    wait_async_le4();
    slot_read(&stage[(Sq - 2) & 3][tid][0], a, x);
    Yprev = c22_mma(a, Yprev, x);
    c22_store_nt(dst + (size_t)(Sq - 2) * step, Yprev);
    wait_async_le0();
    slot_read(&stage[(Sq - 1) & 3][tid][0], a, x);
    Yprev = c22_mma(a, Yprev, x);
    c22_store_nt(dst + (size_t)(Sq - 1) * step, Yprev);
  }
}

// ---------------------------------------------------------------------------
extern "C" void kernel_launch(void* const* d_in, const int* in_sizes, int n_in,
                              void* d_out, int out_size, void* d_ws, size_t ws_size,
                              hipStream_t stream) {
  (void)in_sizes; (void)n_in; (void)out_size; (void)ws_size;
  const float* A = (const float*)d_in[0];
  const float* X = (const float*)d_in[1];
  float* Y = (float*)d_out;

  float* ws  = (float*)d_ws;
  float* agg = ws;                                  // Pq*NCHAIN*16 floats (4 MB)
  float* yin = ws + (size_t)Pq * NCHAIN * 16;       // Pq*NCHAIN*8  floats (2 MB)

  const int gridMain = Bq * Pq * (Cq / NT);         // 256 blocks
  k1_partial<<<gridMain, NT, 0, stream>>>(A, X, agg);
  k2_carry<<<NCHAIN / NT, NT, 0, stream>>>(agg, yin);
  k3_apply<<<gridMain, NT, 0, stream>>>(A, X, yin, Y);
}